// HierarchicalPooling_79242146611960
// MI455X (gfx1250) — compile-verified
//
#include <hip/hip_runtime.h>
#include <math.h>

#define N_NODES 500000
#define H       256
#define G       8192
#define LN_EPS  1e-5f

typedef __attribute__((ext_vector_type(2))) float v2f;
typedef __attribute__((ext_vector_type(4))) float v4f;
typedef __attribute__((ext_vector_type(8))) float v8f;

// Order-preserving encode of f32 into u32 so segment-max can use a single
// native global_atomic_max_u32 (no CAS loop).
__device__ __forceinline__ unsigned enc_f32(float f) {
    unsigned u = __float_as_uint(f);
    return (u & 0x80000000u) ? ~u : (u | 0x80000000u);
}
__device__ __forceinline__ float dec_f32(unsigned k) {
    unsigned u = (k & 0x80000000u) ? (k & 0x7FFFFFFFu) : ~k;
    return __uint_as_float(u);
}

// ---------------------------------------------------------------------------
// Phase 1: streaming scatter-reduce. 512 MB of x reads (HBM-bound, ~22us
// roofline). x is read exactly once -> non-temporal loads so the stream does
// not evict the 16MB of L2-resident atomic targets (sums/maxenc/counts).
// Each thread owns one float4 (4 channels) of one node.
// ---------------------------------------------------------------------------
__global__ __launch_bounds__(256) void scatter_pool_kernel(
    const float* __restrict__ x, const int* __restrict__ fid,
    float* __restrict__ sums, unsigned* __restrict__ maxenc,
    float* __restrict__ counts)
{
    int t = blockIdx.x * 256 + threadIdx.x;          // N*64 = 32M threads exactly
    int node  = t >> 6;
    int chunk = t & 63;
    if (node >= N_NODES) return;
    int g = fid[node];
    const v4f v = __builtin_nontemporal_load((const v4f*)x + t);  // b128, TH=NT
    int base = g * H + chunk * 4;
    atomicAdd(&sums[base + 0], v.x);
    atomicAdd(&sums[base + 1], v.y);
    atomicAdd(&sums[base + 2], v.z);
    atomicAdd(&sums[base + 3], v.w);
    atomicMax(&maxenc[base + 0], enc_f32(v.x));
    atomicMax(&maxenc[base + 1], enc_f32(v.y));
    atomicMax(&maxenc[base + 2], enc_f32(v.z));
    atomicMax(&maxenc[base + 3], enc_f32(v.w));
    if (chunk == 0) atomicAdd(&counts[g], 1.0f);
}

// ---------------------------------------------------------------------------
// Fold Wf[768,256] -> Wsum[256,256]: concat([f,f,f]) @ Wf == f @ (W0+W1+W2).
// ---------------------------------------------------------------------------
__global__ __launch_bounds__(256) void fold_wf_kernel(
    const float* __restrict__ Wf, float* __restrict__ Wsum)
{
    int i = blockIdx.x * 256 + threadIdx.x;          // 0..65535
    Wsum[i] = Wf[i] + Wf[i + 256 * H] + Wf[i + 512 * H];
}

// ---------------------------------------------------------------------------
// Phase 2+3 fused: per block, 16 group-rows end to end.
//   pooled -> (WMMA f32 16x16x4, K=512) -> +bp -> LN1 -> ReLU
//          -> (WMMA f32 16x16x4, K=256 vs folded Wf) -> +bf -> LN2 -> ReLU -> out
// 8 waves; wave w owns output columns [w*32, w*32+32) as two 16x16 N-tiles.
// LDS row strides 516/260 keep the 16 M-lanes on distinct banks (paired K
// reads merge into ds_load_2addr_b64).
// ---------------------------------------------------------------------------
__global__ __launch_bounds__(256) void fused_mlp_kernel(
    const float* __restrict__ sums, const unsigned* __restrict__ maxenc,
    const float* __restrict__ counts,
    const float* __restrict__ Wp, const float* __restrict__ bp,
    const float* __restrict__ g1, const float* __restrict__ b1,
    const float* __restrict__ Wsum, const float* __restrict__ bf,
    const float* __restrict__ g2, const float* __restrict__ b2,
    float* __restrict__ out)
{
    __shared__ float Apool[16 * 516];   // pooled tile, later reused as pre-LN h
    __shared__ float Afunc[16 * 260];   // post-LN1 activations (GEMM2 A)
    __shared__ float redS[256];
    __shared__ float redQ[256];

    const int tid   = threadIdx.x;
    const int lane  = tid & 31;
    const int wave  = tid >> 5;           // 0..7
    const int m     = lane & 15;          // M (A) / N (B) index within tile
    const int khalf = (lane >> 4) << 1;   // 0 or 2: K sub-block per half-wave
    const int rbase = (lane >> 4) << 3;   // C/D rows: 0..7 or 8..15
    const int g0    = blockIdx.x * 16;

    // ---- stage pooled = [mean | max] rows into LDS -------------------------
    for (int i = tid; i < 16 * 512; i += 256) {
        int row = i >> 9, col = i & 511;
        int g = g0 + row;
        float v;
        if (col < 256) v = sums[g * H + col] / fmaxf(counts[g], 1.0f);
        else           v = dec_f32(maxenc[g * H + (col - 256)]);
        Apool[row * 516 + col] = v;
    }
    __syncthreads();

    const int n0 = wave * 32 + m;
    const int n1 = n0 + 16;

    // ---- GEMM1: h = pooled @ Wp  (K = 512, f32 WMMA) -----------------------
    v8f acc0 = {0.f,0.f,0.f,0.f,0.f,0.f,0.f,0.f};
    v8f acc1 = {0.f,0.f,0.f,0.f,0.f,0.f,0.f,0.f};
    for (int kk = 0; kk < 512; kk += 4) {
        int k0 = kk + khalf;
        v2f a;  a.x  = Apool[m * 516 + k0];   a.y  = Apool[m * 516 + k0 + 1];
        v2f b0; b0.x = Wp[k0 * H + n0];       b0.y = Wp[(k0 + 1) * H + n0];
        v2f b1v; b1v.x = Wp[k0 * H + n1];     b1v.y = Wp[(k0 + 1) * H + n1];
        acc0 = __builtin_amdgcn_wmma_f32_16x16x4_f32(false, a, false, b0,
                                                     (short)0, acc0, false, false);
        acc1 = __builtin_amdgcn_wmma_f32_16x16x4_f32(false, a, false, b1v,
                                                     (short)0, acc1, false, false);
    }
    __syncthreads();                    // Apool reads done; reuse as h buffer
    float* hbuf = Apool;                // viewed with stride 260

#pragma unroll
    for (int r = 0; r < 8; ++r) {
        int row = rbase + r;
        hbuf[row * 260 + n0] = acc0[r] + bp[n0];
        hbuf[row * 260 + n1] = acc1[r] + bp[n1];
    }
    __syncthreads();

    // ---- LN1 + ReLU -> Afunc ----------------------------------------------
    {
        int row = tid >> 4, c = tid & 15;
        float s = 0.f, q = 0.f;
        for (int j = c; j < 256; j += 16) {
            float v = hbuf[row * 260 + j]; s += v; q += v * v;
        }
        redS[tid] = s; redQ[tid] = q;
    }
    __syncthreads();
    {
        int row = tid >> 4, c = tid & 15;
        float S = 0.f, Q = 0.f;
#pragma unroll
        for (int j = 0; j < 16; ++j) { S += redS[row * 16 + j]; Q += redQ[row * 16 + j]; }
        float mu   = S * (1.0f / 256.0f);
        float var  = Q * (1.0f / 256.0f) - mu * mu;
        float rstd = rsqrtf(var + LN_EPS);
#pragma unroll
        for (int j = 0; j < 16; ++j) {
            int col = c + j * 16;
            float v = hbuf[row * 260 + col];
            Afunc[row * 260 + col] =
                fmaxf((v - mu) * rstd * g1[col] + b1[col], 0.0f);
        }
    }
    __syncthreads();

    // ---- GEMM2: h2 = func @ Wsum  (K = 256, f32 WMMA) ----------------------
    v8f acc2 = {0.f,0.f,0.f,0.f,0.f,0.f,0.f,0.f};
    v8f acc3 = {0.f,0.f,0.f,0.f,0.f,0.f,0.f,0.f};
    for (int kk = 0; kk < 256; kk += 4) {
        int k0 = kk + khalf;
        v2f a;  a.x  = Afunc[m * 260 + k0];   a.y  = Afunc[m * 260 + k0 + 1];
        v2f b0; b0.x = Wsum[k0 * H + n0];     b0.y = Wsum[(k0 + 1) * H + n0];
        v2f b1v; b1v.x = Wsum[k0 * H + n1];   b1v.y = Wsum[(k0 + 1) * H + n1];
        acc2 = __builtin_amdgcn_wmma_f32_16x16x4_f32(false, a, false, b0,
                                                     (short)0, acc2, false, false);
        acc3 = __builtin_amdgcn_wmma_f32_16x16x4_f32(false, a, false, b1v,
                                                     (short)0, acc3, false, false);
    }
    __syncthreads();

#pragma unroll
    for (int r = 0; r < 8; ++r) {
        int row = rbase + r;
        hbuf[row * 260 + n0] = acc2[r] + bf[n0];
        hbuf[row * 260 + n1] = acc3[r] + bf[n1];
    }
    __syncthreads();

    // ---- LN2 + ReLU -> global out -----------------------------------------
    {
        int row = tid >> 4, c = tid & 15;
        float s = 0.f, q = 0.f;
        for (int j = c; j < 256; j += 16) {
            float v = hbuf[row * 260 + j]; s += v; q += v * v;
        }
        redS[tid] = s; redQ[tid] = q;
    }
    __syncthreads();
    {
        int row = tid >> 4, c = tid & 15;
        float S = 0.f, Q = 0.f;
#pragma unroll
        for (int j = 0; j < 16; ++j) { S += redS[row * 16 + j]; Q += redQ[row * 16 + j]; }
        float mu   = S * (1.0f / 256.0f);
        float var  = Q * (1.0f / 256.0f) - mu * mu;
        float rstd = rsqrtf(var + LN_EPS);
        int g = g0 + row;
#pragma unroll
        for (int j = 0; j < 16; ++j) {
            int col = c + j * 16;
            float v = hbuf[row * 260 + col];
            out[(size_t)g * H + col] =
                fmaxf((v - mu) * rstd * g2[col] + b2[col], 0.0f);
        }
    }
}

// ---------------------------------------------------------------------------
extern "C" void kernel_launch(void* const* d_in, const int* in_sizes, int n_in,
                              void* d_out, int out_size, void* d_ws, size_t ws_size,
                              hipStream_t stream)
{
    const float* x   = (const float*)d_in[0];
    // d_in[1] = batch (unused by reference output path)
    const int*   fid = (const int*)d_in[2];
    const float* Wp  = (const float*)d_in[3];
    const float* bp  = (const float*)d_in[4];
    const float* g1  = (const float*)d_in[5];
    const float* b1  = (const float*)d_in[6];
    const float* Wf  = (const float*)d_in[7];
    const float* bf  = (const float*)d_in[8];
    const float* g2  = (const float*)d_in[9];
    const float* b2  = (const float*)d_in[10];
    float* out = (float*)d_out;

    // workspace layout (16.3 MB): sums | maxenc | counts | Wsum
    float*    sums   = (float*)d_ws;
    unsigned* maxenc = (unsigned*)(sums + (size_t)G * H);
    float*    counts = (float*)(maxenc + (size_t)G * H);
    float*    Wsum   = counts + G;

    // zero sums+maxenc+counts (maxenc==0 decodes below any real float's key)
    hipMemsetAsync(d_ws, 0, ((size_t)2 * G * H + G) * sizeof(float), stream);

    int blocksB = (N_NODES * 64) / 256;   // 125000, exact
    scatter_pool_kernel<<<blocksB, 256, 0, stream>>>(x, fid, sums, maxenc, counts);
    fold_wf_kernel<<<(H * H) / 256, 256, 0, stream>>>(Wf, Wsum);
    fused_mlp_kernel<<<G / 16, 256, 0, stream>>>(sums, maxenc, counts,
                                                 Wp, bp, g1, b1,
                                                 Wsum, bf, g2, b2, out);
}